// SubsetOperator_88407606821498
// MI455X (gfx1250) — compile-verified
//
#include <hip/hip_runtime.h>

typedef float f4  __attribute__((ext_vector_type(4)));
typedef float v2f __attribute__((ext_vector_type(2)));
typedef float v8f __attribute__((ext_vector_type(8)));

#define N_COLS   16384
#define THREADS  256
#define NWAVES   (THREADS / 32)
#define CHUNKS   (N_COLS / (THREADS * 4))   // 16 float4 chunks per thread
#define K_ITERS  4
#define EPSILON  1.1754943508222875e-38f    // np.finfo(float32).tiny
#define NEG_INF  (-3.402823466e+38f)

// Wave32 sum-reduction via V_WMMA_F32_16X16X4_F32 with an all-ones B matrix.
// A layout (16x4 f32): vgpr0 = lanes' first value (K0 / K2), vgpr1 = second (K1 / K3).
// With a = {ts, 0}: rowsum[m] = ts[m] + ts[m+16]. D[m][n] = rowsum[m] for all n.
// Lanes 0-15 hold rowsum[0..7] in d[0..7]; lanes 16-31 hold rowsum[8..15].
__device__ __forceinline__ float wave_sum_wmma(float ts) {
    v2f a; a[0] = ts;  a[1] = 0.0f;
    v2f b; b[0] = 1.0f; b[1] = 1.0f;
    v8f c = {};
    v8f d = __builtin_amdgcn_wmma_f32_16x16x4_f32(
        /*neg_a=*/false, a, /*neg_b=*/false, b,
        /*c_mod=*/(short)0, c, /*reuse_a=*/false, /*reuse_b=*/false);
    float ws = d[0] + d[1] + d[2] + d[3] + d[4] + d[5] + d[6] + d[7];
    ws += __shfl_xor(ws, 16, 32);   // combine the two half-wave row groups
    return ws;                      // full 32-lane sum, uniform across the wave
}

__device__ __forceinline__ float wave_max(float m) {
    #pragma unroll
    for (int off = 16; off > 0; off >>= 1)
        m = fmaxf(m, __shfl_xor(m, off, 32));
    return m;
}

__global__ __launch_bounds__(THREADS, 1)
void subset_op_kernel(const float* __restrict__ scores,
                      const float* __restrict__ g,
                      float* __restrict__ out) {
    extern __shared__ float smem[];
    float* s_row = smem;               // N_COLS floats: running logits (then exp values)
    float* k_row = smem + N_COLS;      // N_COLS floats: khot accumulator
    float* red   = smem + 2 * N_COLS;  // [0..7] max scratch, [8..15] sum scratch

    const int t    = threadIdx.x;
    const int lane = t & 31;
    const int wave = t >> 5;
    const int row  = blockIdx.x;

    const float* sc = scores + (size_t)row * N_COLS;
    const float* gg = g      + (size_t)row * N_COLS;

    // ---- stage s0 = scores + g into LDS (read inputs exactly once, NT), zero khot ----
    const f4 fzero = {0.f, 0.f, 0.f, 0.f};
    #pragma unroll
    for (int i = 0; i < CHUNKS; ++i) {
        const int idx = i * (THREADS * 4) + t * 4;
        f4 a = __builtin_nontemporal_load((const f4*)(sc + idx));
        f4 b = __builtin_nontemporal_load((const f4*)(gg + idx));
        *(f4*)(s_row + idx) = a + b;
        *(f4*)(k_row + idx) = fzero;
    }
    __syncthreads();

    for (int it = 0; it < K_ITERS; ++it) {
        // ---- row max (shuffle tree within wave, LDS across waves) ----
        float m = NEG_INF;
        #pragma unroll
        for (int i = 0; i < CHUNKS; ++i) {
            const int idx = i * (THREADS * 4) + t * 4;
            f4 v = *(const f4*)(s_row + idx);
            m = fmaxf(m, fmaxf(fmaxf(v.x, v.y), fmaxf(v.z, v.w)));
        }
        m = wave_max(m);
        if (lane == 0) red[wave] = m;
        __syncthreads();
        m = red[0];
        #pragma unroll
        for (int w = 1; w < NWAVES; ++w) m = fmaxf(m, red[w]);

        // ---- e = exp(s - m) stored in place; accumulate partial sum ----
        float ts = 0.f;
        #pragma unroll
        for (int i = 0; i < CHUNKS; ++i) {
            const int idx = i * (THREADS * 4) + t * 4;
            f4 v = *(const f4*)(s_row + idx);
            f4 e;
            e.x = __expf(v.x - m); e.y = __expf(v.y - m);
            e.z = __expf(v.z - m); e.w = __expf(v.w - m);
            *(f4*)(s_row + idx) = e;
            ts += (e.x + e.y) + (e.z + e.w);
        }
        // wave sum on the matrix pipe, then cross-wave via LDS
        float ws = wave_sum_wmma(ts);
        if (lane == 0) red[8 + wave] = ws;
        __syncthreads();
        float Z = 0.f;
        #pragma unroll
        for (int w = 0; w < NWAVES; ++w) Z += red[8 + w];
        const float invZ = __builtin_amdgcn_rcpf(Z);

        // ---- p = e/Z; khot += p; s = m + log(e * max(1-p, eps)) (fused single log) ----
        #pragma unroll
        for (int i = 0; i < CHUNKS; ++i) {
            const int idx = i * (THREADS * 4) + t * 4;
            f4 e  = *(const f4*)(s_row + idx);
            f4 p  = e * invZ;
            f4 kh = *(const f4*)(k_row + idx);
            kh += p;
            *(f4*)(k_row + idx) = kh;
            f4 sn;
            sn.x = m + __logf(e.x * fmaxf(1.f - p.x, EPSILON));
            sn.y = m + __logf(e.y * fmaxf(1.f - p.y, EPSILON));
            sn.z = m + __logf(e.z * fmaxf(1.f - p.z, EPSILON));
            sn.w = m + __logf(e.w * fmaxf(1.f - p.w, EPSILON));
            *(f4*)(s_row + idx) = sn;
        }
        __syncthreads();
    }

    // ---- stream khot out K times (output > L2, never re-read -> NT b128 stores) ----
    float* outrow = out + (size_t)row * (size_t)(K_ITERS * N_COLS);
    #pragma unroll
    for (int i = 0; i < CHUNKS; ++i) {
        const int idx = i * (THREADS * 4) + t * 4;
        f4 v = *(const f4*)(k_row + idx);
        #pragma unroll
        for (int kk = 0; kk < K_ITERS; ++kk)
            __builtin_nontemporal_store(v, (f4*)(outrow + kk * N_COLS + idx));
    }
}

extern "C" void kernel_launch(void* const* d_in, const int* in_sizes, int n_in,
                              void* d_out, int out_size, void* d_ws, size_t ws_size,
                              hipStream_t stream) {
    const float* scores = (const float*)d_in[0];
    const float* g      = (const float*)d_in[1];
    float* out          = (float*)d_out;

    const int B = in_sizes[0] / N_COLS;                     // 2048 rows
    const size_t shmem = (size_t)(2 * N_COLS + 32) * sizeof(float);  // 128 KB + scratch (<320 KB/WGP)
    subset_op_kernel<<<dim3(B), dim3(THREADS), shmem, stream>>>(scores, g, out);
}